// BNs_7902739824826
// MI455X (gfx1250) — compile-verified
//
#include <hip/hip_runtime.h>
#include <hip/hip_bf16.h>

typedef __attribute__((ext_vector_type(2))) float v2f;
typedef __attribute__((ext_vector_type(4))) float f32x4;
typedef __attribute__((ext_vector_type(8))) float v8f;

static constexpr int   Bdim   = 32;
static constexpr int   Cdim   = 64;
static constexpr int   HW     = 1024;             // 32*32
static constexpr int   Nbr    = 16;               // branches
static constexpr int   XSTRB  = Cdim * HW;        // 65536 floats per batch image
static constexpr float EPSF   = 1e-5f;
static constexpr float INV_CNT = 1.0f / (float)(Bdim * HW);  // 1/32768

// ---------------------------------------------------------------------------
// Pass 1: per-channel sum / sum-of-squares -> mean, rsqrt(var+eps)
// grid = 64 (one block per channel), block = 256 (8 waves)
// ---------------------------------------------------------------------------
__global__ __launch_bounds__(256) void bn_stats_kernel(const float* __restrict__ x,
                                                       float* __restrict__ stats) {
  const int c = blockIdx.x;     // channel
  const int t = threadIdx.x;    // 0..255

  float s = 0.0f, q = 0.0f;
  const float* base = x + (size_t)c * HW + (size_t)t * 4;
#pragma unroll 8
  for (int b = 0; b < Bdim; ++b) {
    const f32x4 v = *reinterpret_cast<const f32x4*>(base + (size_t)b * XSTRB);
    s += v.x + v.y + v.z + v.w;
    q = fmaf(v.x, v.x, q);
    q = fmaf(v.y, v.y, q);
    q = fmaf(v.z, v.z, q);
    q = fmaf(v.w, v.w, q);
  }

  __shared__ float ls[256];
  __shared__ float lq[256];
  ls[t] = s;
  lq[t] = q;
  __syncthreads();

  if (t < 32) {  // wave 0 only: EXEC all-ones within this wave (WMMA requirement)
    float ts = 0.0f, tq = 0.0f;
#pragma unroll
    for (int w = 0; w < 8; ++w) {
      ts += ls[t + 32 * w];
      tq += lq[t + 32 * w];
    }

    // Cross-lane reduce 32 -> 16 via V_WMMA_F32_16X16X4_F32 with A = ones:
    //   D[m,n] = sum_k B[k,n].  B f32 layout: VGPR0 holds K=0 (lanes 0-15) and
    //   K=2 (lanes 16-31), VGPR1 holds K=1/K=3.  b = {partial, 0} per lane
    //   => D[m,n] = partial_n + partial_{n+16}, replicated over rows m.
    v8f accs = {};
    v8f accq = {};
    const v2f ones = {1.0f, 1.0f};
    v2f bs;  bs.x = ts;  bs.y = 0.0f;
    v2f bq;  bq.x = tq;  bq.y = 0.0f;
    accs = __builtin_amdgcn_wmma_f32_16x16x4_f32(false, ones, false, bs,
                                                 (short)0, accs, false, false);
    accq = __builtin_amdgcn_wmma_f32_16x16x4_f32(false, ones, false, bq,
                                                 (short)0, accq, false, false);

    float rs = accs[0];   // lanes 0-15: N=0..15 (lanes 16-31 are row M=8 copies)
    float rq = accq[0];
    rs += __shfl_xor(rs, 8, 32);  rq += __shfl_xor(rq, 8, 32);
    rs += __shfl_xor(rs, 4, 32);  rq += __shfl_xor(rq, 4, 32);
    rs += __shfl_xor(rs, 2, 32);  rq += __shfl_xor(rq, 2, 32);
    rs += __shfl_xor(rs, 1, 32);  rq += __shfl_xor(rq, 1, 32);

    if (t == 0) {
      const float mean = rs * INV_CNT;
      const float var  = fmaf(-mean, mean, rq * INV_CNT);  // E[x^2] - mean^2
      stats[c]        = mean;
      stats[Cdim + c] = rsqrtf(var + EPSF);
    }
  }
}

// ---------------------------------------------------------------------------
// Pass 2: normalize once, fan out 16 affine branches.
// grid = B*C = 2048 (one (b,c) 1024-elem plane per block), block = 256
// x is 8 MiB -> guaranteed L2-resident after pass 1; output is 128 MiB
// streaming write-once -> non-temporal stores.
// ---------------------------------------------------------------------------
__global__ __launch_bounds__(256) void bn_apply_kernel(const float* __restrict__ x,
                                                       const float* __restrict__ gamma,
                                                       const float* __restrict__ beta,
                                                       const float* __restrict__ stats,
                                                       float* __restrict__ out) {
  const int plane = blockIdx.x;        // b*C + c
  const int c = plane & (Cdim - 1);
  const int b = plane >> 6;
  const int t = threadIdx.x;

  __shared__ float sg[Nbr];
  __shared__ float sb[Nbr];
  if (t < Nbr) {
    sg[t] = gamma[t * Cdim + c];
    sb[t] = beta[t * Cdim + c];
  }
  const float mean = stats[c];         // block-uniform -> scalar load
  const float inv  = stats[Cdim + c];
  __syncthreads();

  const f32x4 v =
      *reinterpret_cast<const f32x4*>(x + (size_t)plane * HW + (size_t)t * 4);
  f32x4 xh;
  xh.x = (v.x - mean) * inv;
  xh.y = (v.y - mean) * inv;
  xh.z = (v.z - mean) * inv;
  xh.w = (v.w - mean) * inv;

  float* ob = out + (size_t)b * ((size_t)Nbr * Cdim * HW)
                  + (size_t)c * HW + (size_t)t * 4;
#pragma unroll
  for (int n = 0; n < Nbr; ++n) {
    const float g  = sg[n];
    const float be = sb[n];
    f32x4 o;
    o.x = fmaf(g, xh.x, be);
    o.y = fmaf(g, xh.y, be);
    o.z = fmaf(g, xh.z, be);
    o.w = fmaf(g, xh.w, be);
    __builtin_nontemporal_store(
        o, reinterpret_cast<f32x4*>(ob + (size_t)n * (Cdim * HW)));
  }
}

// ---------------------------------------------------------------------------
extern "C" void kernel_launch(void* const* d_in, const int* in_sizes, int n_in,
                              void* d_out, int out_size, void* d_ws, size_t ws_size,
                              hipStream_t stream) {
  const float* x     = (const float*)d_in[0];  // [32,64,32,32]
  const float* gamma = (const float*)d_in[1];  // [16,64]
  const float* beta  = (const float*)d_in[2];  // [16,64]
  float* out   = (float*)d_out;                // [32,1024,32,32]
  float* stats = (float*)d_ws;                 // [2,64]: mean, inv

  bn_stats_kernel<<<Cdim, 256, 0, stream>>>(x, stats);
  bn_apply_kernel<<<Bdim * Cdim, 256, 0, stream>>>(x, gamma, beta, stats, out);
}